// ModulatedDeformableConv2d_40149354283832
// MI455X (gfx1250) — compile-verified
//
#include <hip/hip_runtime.h>
#include <hip/hip_bf16.h>
#include <math.h>

// ---------------- problem constants (from reference) ----------------
#define B_    4
#define C_INN 64
#define C_OUTT 64
#define C_OFF 32
#define H_    160
#define W_    160
#define HW    (H_ * W_)          // 25600
#define KK_   9
#define KTOT  (C_INN * KK_)      // 576  (GEMM K dimension, index = c*9 + kk)

typedef float v2f __attribute__((ext_vector_type(2)));
typedef float v8f __attribute__((ext_vector_type(8)));
typedef unsigned int v4u __attribute__((ext_vector_type(4)));
typedef int v4i __attribute__((ext_vector_type(4)));
typedef int v8i __attribute__((ext_vector_type(8)));

// =====================================================================
// Kernel 0: transpose x [B, C, HW] -> x_t [B, HW, C]
// so that a bilinear corner read touches 64 contiguous channels (256 B).
// =====================================================================
__global__ __launch_bounds__(256) void k_transpose(const float* __restrict__ x,
                                                   float* __restrict__ xt) {
    __shared__ float tile[32][33];
    int blk = blockIdx.x;
    int pt = blk % (HW / 32);  blk /= (HW / 32);
    int ct = blk % (C_INN / 32);
    int b  = blk / (C_INN / 32);
    int p0 = pt * 32, c0 = ct * 32;
    int tx = threadIdx.x & 31, ty = threadIdx.x >> 5;

    const float* xb  = x  + (size_t)b * C_INN * HW;
    float*       xtb = xt + (size_t)b * HW * C_INN;

#pragma unroll
    for (int i = 0; i < 4; ++i) {
        int c = ty + i * 8;
        tile[c][tx] = xb[(size_t)(c0 + c) * HW + p0 + tx];
    }
    __syncthreads();
#pragma unroll
    for (int i = 0; i < 4; ++i) {
        int r = ty + i * 8;
        xtb[(size_t)(p0 + r) * C_INN + c0 + tx] = tile[tx][r];
    }
}

// =====================================================================
// Kernel 1: offset conv  om[b][27][h][w] = conv3x3(feat, w_off) + b_off
// =====================================================================
__global__ __launch_bounds__(256) void k_offconv(const float* __restrict__ feat,
                                                 const float* __restrict__ w_off,
                                                 const float* __restrict__ b_off,
                                                 float* __restrict__ om) {
    int gid = blockIdx.x * 256 + threadIdx.x;
    const int TOT = B_ * 27 * HW;
    if (gid >= TOT) return;
    int p = gid % HW;
    int t = gid / HW;
    int j = t % 27;
    int b = t / 27;
    int h = p / W_, w = p % W_;

    float acc = b_off[j];
    const float* fb = feat + (size_t)b * C_OFF * HW;
    const float* wj = w_off + (size_t)j * C_OFF * 9;

    for (int ci = 0; ci < C_OFF; ++ci) {
        const float* fc = fb + (size_t)ci * HW;
        const float* wc = wj + ci * 9;
#pragma unroll
        for (int ky = 0; ky < 3; ++ky) {
            int hy = h + ky - 1;
            if ((unsigned)hy < (unsigned)H_) {
#pragma unroll
                for (int kx = 0; kx < 3; ++kx) {
                    int wx = w + kx - 1;
                    if ((unsigned)wx < (unsigned)W_)
                        acc = fmaf(fc[hy * W_ + wx], wc[ky * 3 + kx], acc);
                }
            }
        }
    }
    om[gid] = acc;
}

// =====================================================================
// Kernel 2: fused bilinear-gather + modulated GEMM via WMMA f32 16x16x4
//  - TDM (tensor_load_to_lds) DMAs the 64x576 weight panel into LDS,
//    overlapped with offset-decode + gather; pad_interval=64DW/amount=2DW
//    makes the LDS row stride 594 words (18 mod 64 -> conflict-free A reads)
//  - B-panel s_lds[K=576][32+pad] built by bilinear gather, mask folded in
//  - out tile: M=64 (all C_OUT) x N=32 pixels; wave (wm,wn) owns 16x16
// =====================================================================
#define SPAD    40                              // 32 cols padded (bank spread)
#define WROW    594                             // 576 + 9 pads * 2 DW (TDM padding)
#define OFF_S   0
#define OFF_CW  (KTOT * SPAD)                   // 23040
#define OFF_IC  (OFF_CW + 288 * 4)              // 24192
#define OFF_W   (OFF_IC + 288 * 4)              // 25344
#define SMEM_FLOATS (OFF_W + C_OUTT * WROW)     // 25344 + 38016 = 63360 fl = 253440 B

__global__ __launch_bounds__(256) void k_mdcn_gemm(const float* __restrict__ xt,
                                                   const float* __restrict__ om,
                                                   const float* __restrict__ weight,
                                                   const float* __restrict__ bias,
                                                   float* __restrict__ out) {
    extern __shared__ float smem[];
    float* s_lds = smem + OFF_S;          // [KTOT][SPAD] B-panel
    float* cwt   = smem + OFF_CW;         // [288][4] corner weights (mask folded)
    int*   ict   = (int*)(smem + OFF_IC); // [288][4] y0c,y1c,x0c,x1c
    float* w_lds = smem + OFF_W;          // [64][WROW] A-panel (TDM-filled)

    const int tid  = threadIdx.x;
    const int tile = blockIdx.x % (HW / 32);
    const int b    = blockIdx.x / (HW / 32);
    const int p0   = tile * 32;

    // ---- Kick off TDM: weight[64][576] --> w_lds with per-64DW padding ----
    if ((tid >> 5) == 0) {
        unsigned long long ga = (unsigned long long)(const void*)weight;
        unsigned int lds_addr = __builtin_amdgcn_groupstaticsize() + OFF_W * 4u;

        v4u g0;
        g0.x = 1u;                                   // count=1 valid descriptor
        g0.y = lds_addr;                             // LDS byte address
        g0.z = (unsigned int)(ga & 0xFFFFFFFFull);   // global_addr[31:0]
        g0.w = (unsigned int)((ga >> 32) & 0x1FFFFFFull) | (2u << 30); // [56:32] | type=2

        v8i g1;
        g1[0] = (int)((1u << 25) | (5u << 22) | (1u << 20) | (2u << 16));
                 // pad_amount=2DW(code1), pad_interval=64DW(code5), pad_en, data_size=4B
        g1[1] = (int)(576u << 16);                   // tensor_dim0[15:0]=576 in [31:16]
        g1[2] = (int)(64u << 16);                    // tensor_dim1[15:0]=64 in [31:16]
        g1[3] = (int)(576u << 16);                   // tile_dim0=576 in [31:16]
        g1[4] = 64;                                  // tile_dim1=64, tile_dim2=0
        g1[5] = 576;                                 // tensor_dim0_stride[31:0]
        g1[6] = 0;
        g1[7] = 0;

        v4i gz = {0, 0, 0, 0};                       // 2-D tile: groups 2/3 unused
#if defined(__clang_major__) && (__clang_major__ >= 23)
        v8i gz8 = {0, 0, 0, 0, 0, 0, 0, 0};
        __builtin_amdgcn_tensor_load_to_lds(g0, g1, gz, gz, gz8, 0);
#else
        __builtin_amdgcn_tensor_load_to_lds(g0, g1, gz, gz, 0);
#endif
    }

    // ---- Phase A: per (tap, pixel) sampling coords + weights --------
    for (int idx = tid; idx < 288; idx += 256) {
        int kk  = idx >> 5, pix = idx & 31;
        int p   = p0 + pix;
        int h   = p / W_, w = p % W_;
        const float* omb = om + (size_t)b * 27 * HW + p;
        float dy = omb[(size_t)kk * HW];
        float dx = omb[(size_t)(9 + kk) * HW];
        float mz = omb[(size_t)(18 + kk) * HW];
        float m  = 1.0f / (1.0f + __expf(-mz));     // sigmoid mask

        float py = dy + (float)(kk / 3) + (float)(h - 1);   // stride=1, pad=1
        float px = dx + (float)(kk % 3) + (float)(w - 1);
        float y0f = floorf(py), x0f = floorf(px);
        float wy1 = py - y0f, wy0 = 1.0f - wy1;
        float wx1 = px - x0f, wx0 = 1.0f - wx1;
        int y0 = (int)y0f, x0 = (int)x0f;
        int y1 = y0 + 1,   x1 = x0 + 1;
        bool vy0 = (unsigned)y0 < (unsigned)H_, vy1 = (unsigned)y1 < (unsigned)H_;
        bool vx0 = (unsigned)x0 < (unsigned)W_, vx1 = (unsigned)x1 < (unsigned)W_;

        cwt[idx * 4 + 0] = (vy0 && vx0) ? wy0 * wx0 * m : 0.0f;
        cwt[idx * 4 + 1] = (vy0 && vx1) ? wy0 * wx1 * m : 0.0f;
        cwt[idx * 4 + 2] = (vy1 && vx0) ? wy1 * wx0 * m : 0.0f;
        cwt[idx * 4 + 3] = (vy1 && vx1) ? wy1 * wx1 * m : 0.0f;
        ict[idx * 4 + 0] = min(max(y0, 0), H_ - 1);
        ict[idx * 4 + 1] = min(max(y1, 0), H_ - 1);
        ict[idx * 4 + 2] = min(max(x0, 0), W_ - 1);
        ict[idx * 4 + 3] = min(max(x1, 0), W_ - 1);
    }
    __syncthreads();

    // ---- Phase B: gather -> B-panel in LDS --------------------------
    // wave per (tap,pixel) task; lane covers channels 2*lane, 2*lane+1.
    {
        const int wave = tid >> 5, lane = tid & 31;
        const float* xb = xt + (size_t)b * HW * C_INN;
        for (int task = wave; task < 288; task += 8) {   // 36 uniform iters
            int kk = task >> 5, pix = task & 31;
            float w00 = cwt[task * 4 + 0], w01 = cwt[task * 4 + 1];
            float w10 = cwt[task * 4 + 2], w11 = cwt[task * 4 + 3];
            int y0 = ict[task * 4 + 0], y1 = ict[task * 4 + 1];
            int x0 = ict[task * 4 + 2], x1 = ict[task * 4 + 3];

            const float2* r00 = (const float2*)(xb + (size_t)(y0 * W_ + x0) * C_INN) + lane;
            const float2* r01 = (const float2*)(xb + (size_t)(y0 * W_ + x1) * C_INN) + lane;
            const float2* r10 = (const float2*)(xb + (size_t)(y1 * W_ + x0) * C_INN) + lane;
            const float2* r11 = (const float2*)(xb + (size_t)(y1 * W_ + x1) * C_INN) + lane;
            float2 v00 = *r00, v01 = *r01, v10 = *r10, v11 = *r11;

            float sx = w00 * v00.x + w01 * v01.x + w10 * v10.x + w11 * v11.x;
            float sy = w00 * v00.y + w01 * v01.y + w10 * v10.y + w11 * v11.y;
            int c = lane * 2;
            s_lds[((size_t)(c * KK_ + kk)) * SPAD + pix]       = sx;
            s_lds[((size_t)((c + 1) * KK_ + kk)) * SPAD + pix] = sy;
        }
    }
    // weight DMA must have landed before the GEMM reads w_lds
    __builtin_amdgcn_s_wait_tensorcnt(0);
    __syncthreads();

    // ---- Phase C: 64x32 tile GEMM, K=576 via V_WMMA_F32_16X16X4_F32 --
    const int lane   = tid & 31;
    const int waveId = tid >> 5;
    const int wm     = waveId & 3;       // M tile: rows wm*16..
    const int wn     = waveId >> 2;      // N tile: cols wn*16..
    const int mlane  = lane & 15;
    const int hi     = lane >> 4;        // lanes 16..31 hold K+2 / M+8 halves
    const int col    = wn * 16 + mlane;

    // A from LDS: row stride WROW=594 words; padded col offset = k + 2*(k/64)
    const float* abase = w_lds + (size_t)(wm * 16 + mlane) * WROW + 2 * hi;
    const float* brow0 = &s_lds[(size_t)(2 * hi) * SPAD + col];

    v8f acc = {0.f, 0.f, 0.f, 0.f, 0.f, 0.f, 0.f, 0.f};
#pragma unroll 8
    for (int kb = 0; kb < 144; ++kb) {
        int k0   = kb * 4;
        int kadj = k0 + ((kb >> 4) << 1);            // TDM pad: +2 DW per 64 DW
        v2f a = *(const v2f*)(abase + kadj);         // {W[m][k], W[m][k+1]}
        v2f bb;
        bb.x = brow0[(size_t)k0 * SPAD];             // s[k0+2hi][col]
        bb.y = brow0[(size_t)(k0 + 1) * SPAD];       // s[k0+1+2hi][col]
        acc = __builtin_amdgcn_wmma_f32_16x16x4_f32(
            /*neg_a=*/false, a, /*neg_b=*/false, bb,
            /*c_mod=*/(short)0, acc, /*reuse_a=*/false, /*reuse_b=*/false);
    }

    // ---- Epilogue: C/D layout -> out[b][o][p] + bias ----------------
    float* ob = out + (size_t)b * C_OUTT * HW + p0 + col;
#pragma unroll
    for (int r = 0; r < 8; ++r) {
        int o = wm * 16 + hi * 8 + r;                // VGPR r: M=r / M=8+r
        ob[(size_t)o * HW] = acc[r] + bias[o];
    }
}

// =====================================================================
extern "C" void kernel_launch(void* const* d_in, const int* in_sizes, int n_in,
                              void* d_out, int out_size, void* d_ws, size_t ws_size,
                              hipStream_t stream) {
    const float* x      = (const float*)d_in[0];
    const float* feat   = (const float*)d_in[1];
    const float* w_off  = (const float*)d_in[2];
    const float* b_off  = (const float*)d_in[3];
    const float* weight = (const float*)d_in[4];
    const float* bias   = (const float*)d_in[5];
    float*       out    = (float*)d_out;

    // workspace layout: x_t (26.2 MB) | om (11.1 MB)
    float* xt = (float*)d_ws;
    float* om = xt + (size_t)B_ * HW * C_INN;

    k_transpose<<<B_ * (C_INN / 32) * (HW / 32), 256, 0, stream>>>(x, xt);
    k_offconv<<<(B_ * 27 * HW + 255) / 256, 256, 0, stream>>>(feat, w_off, b_off, om);

    size_t smem = (size_t)SMEM_FLOATS * sizeof(float);   // 253440 B
    k_mdcn_gemm<<<B_ * (HW / 32), 256, smem, stream>>>(xt, om, weight, bias, out);
}